// mLSTMCell_21380347199692
// MI455X (gfx1250) — compile-verified
//
#include <hip/hip_runtime.h>
#include <math.h>

#define BB 8
#define TT 512
#define DD 256
#define UU 256

typedef __attribute__((ext_vector_type(2))) float v2f;
typedef __attribute__((ext_vector_type(8))) float v8f;

__device__ __forceinline__ float wave_reduce_sum(float v) {
#pragma unroll
  for (int off = 16; off >= 1; off >>= 1) v += __shfl_xor(v, off, 32);
  return v;
}

// ---------------- LayerNorm: one wave32 per row of x[B*T, D] ----------------
__global__ __launch_bounds__(256) void ln_kernel(const float* __restrict__ x,
                                                 const float* __restrict__ gamma,
                                                 const float* __restrict__ beta,
                                                 float* __restrict__ xn) {
  int row = blockIdx.x * 8 + (threadIdx.x >> 5);
  int l = threadIdx.x & 31;
  const float* xr = x + (size_t)row * DD;
  float vals[8];
  float s = 0.f;
#pragma unroll
  for (int j = 0; j < 8; ++j) { vals[j] = xr[l + 32 * j]; s += vals[j]; }
  s = wave_reduce_sum(s);
  float mu = s * (1.0f / DD);
  float s2 = 0.f;
#pragma unroll
  for (int j = 0; j < 8; ++j) { float d = vals[j] - mu; s2 += d * d; }
  s2 = wave_reduce_sum(s2);
  float rs = rsqrtf(s2 * (1.0f / DD) + 1e-6f);
  float* xo = xn + (size_t)row * DD;
#pragma unroll
  for (int j = 0; j < 8; ++j) {
    int c = l + 32 * j;
    xo[c] = (vals[j] - mu) * rs * gamma[c] + beta[c];
  }
}

// ------------- WMMA fp32 GEMM: out[4096,256] = A[4096,256]*W[256,256]+bias, *scale -------------
// 256 threads = 8 waves; block tile 32(M) x 128(N); wave grid 2x4; TWO 16x16 WMMA tiles per wave
// (A fragment reused). LDS tiles stored K-pair interleaved ([k/2][col][2]) so every WMMA source
// fragment is one aligned ds_load_b64 straight into an even VGPR pair (no repack movs).
__global__ __launch_bounds__(256) void gemm_kernel(const float* __restrict__ A,
                                                   const float* __restrict__ W,
                                                   const float* __restrict__ bias,
                                                   float* __restrict__ out, float scale) {
  __shared__ __align__(16) float As2[32 * 66];   // [kh][m*2 + k&1], kh = k/2 (chunk K=64)
  __shared__ __align__(16) float Bs2[32 * 258];  // [kh][n*2 + k&1], n tile width 128
  const int tid = threadIdx.x;
  const int wave = tid >> 5, lane = tid & 31;
  const int wm = wave >> 2, wn = wave & 3;
  const int lh = lane >> 4, lm = lane & 15;
  const int mBase = blockIdx.y * 32;
  const int nBase = blockIdx.x * 128;

  v8f acc0 = {};
  v8f acc1 = {};
  for (int kc = 0; kc < DD; kc += 64) {
    // Stage A 32(rows) x 64(k): 512 float4 slots, 2 per thread, coalesced.
#pragma unroll
    for (int r = 0; r < 2; ++r) {
      int idx = tid + 256 * r;         // 0..511
      int row = idx >> 4;              // 0..31
      int k4 = (idx & 15) * 4;         // 0..60 (even pairs)
      float4 v = *(const float4*)(A + (size_t)(mBase + row) * DD + kc + k4);
      int kh = k4 >> 1;
      *(float2*)&As2[(kh + 0) * 66 + row * 2] = make_float2(v.x, v.y);
      *(float2*)&As2[(kh + 1) * 66 + row * 2] = make_float2(v.z, v.w);
    }
    // Stage B 64(k) x 128(n): 2048 float4 slots, 8 per thread, coalesced.
#pragma unroll
    for (int r = 0; r < 8; ++r) {
      int idx = tid + 256 * r;         // 0..2047
      int krow = idx >> 5;             // 0..63
      int n4 = (idx & 31) * 4;         // 0..124
      float4 v = *(const float4*)(W + (size_t)(kc + krow) * UU + nBase + n4);
      int kh = krow >> 1, kb = krow & 1;
      Bs2[kh * 258 + (n4 + 0) * 2 + kb] = v.x;
      Bs2[kh * 258 + (n4 + 1) * 2 + kb] = v.y;
      Bs2[kh * 258 + (n4 + 2) * 2 + kb] = v.z;
      Bs2[kh * 258 + (n4 + 3) * 2 + kb] = v.w;
    }
    __syncthreads();
#pragma unroll
    for (int kk = 0; kk < 16; ++kk) {
      int kh = kk * 2 + lh;  // lane half selects K pair (K = 2*kh, 2*kh+1)
      v2f a  = *(const v2f*)&As2[kh * 66 + (wm * 16 + lm) * 2];
      v2f b0 = *(const v2f*)&Bs2[kh * 258 + (wn * 16 + lm) * 2];
      v2f b1 = *(const v2f*)&Bs2[kh * 258 + (64 + wn * 16 + lm) * 2];
      acc0 = __builtin_amdgcn_wmma_f32_16x16x4_f32(false, a, false, b0, (short)0, acc0,
                                                   false, false);
      acc1 = __builtin_amdgcn_wmma_f32_16x16x4_f32(false, a, false, b1, (short)0, acc1,
                                                   false, false);
    }
    __syncthreads();
  }
  // D frag: vgpr v -> M = v + 8*lh, N = lm
  int col0 = nBase + wn * 16 + lm;
  int col1 = col0 + 64;
  float bv0 = bias[col0];
  float bv1 = bias[col1];
#pragma unroll
  for (int v = 0; v < 8; ++v) {
    int row = mBase + wm * 16 + v + 8 * lh;
    out[(size_t)row * UU + col0] = (acc0[v] + bv0) * scale;
    out[(size_t)row * UU + col1] = (acc1[v] + bv1) * scale;
  }
}

// ---------------- Elementwise m-scan + gate nonlinearities (in place) ----------------
__global__ __launch_bounds__(256) void gate_kernel(float* __restrict__ i_t,
                                                   float* __restrict__ f_t) {
  int g = blockIdx.x * 256 + threadIdx.x;  // 0..2047 = (b,u)
  int b = g >> 8, u = g & 255;
  float m = 0.f;  // m0 = 0
  size_t base = (size_t)b * TT * UU + u;
  for (int t = 0; t < TT; ++t) {
    size_t idx = base + (size_t)t * UU;
    float it = i_t[idx];
    m = fmaxf(it + m, it);
    i_t[idx] = __expf(it - m);
    float ft = f_t[idx];
    f_t[idx] = 1.0f / (1.0f + __expf(-ft));
  }
}

// ---------------- Recurrence: one block per batch, C fully register-resident ----------------
// 1024 thr = 32 waves. Wave w owns rows {w+32i}, lane l owns cols {l+32j}; 8x8 regs/thread.
__global__ __launch_bounds__(1024) void recur_kernel(
    const float* __restrict__ qg, const float* __restrict__ kg,
    const float* __restrict__ vg, const float* __restrict__ ig,
    const float* __restrict__ fg, const float* __restrict__ og,
    float* __restrict__ out) {
  __shared__ float f_s[256], a_s[256], k_s[256], q_s[256], o_s[256];
  __shared__ float n_s[256], nq_s[256], num_s[256];
  __shared__ float denom_s;
  const int tid = threadIdx.x;
  const int w = tid >> 5, l = tid & 31;
  const int b = blockIdx.x;

  float C[8][8];
#pragma unroll
  for (int i = 0; i < 8; ++i)
#pragma unroll
    for (int j = 0; j < 8; ++j) C[i][j] = 0.f;
  if (tid < 256) n_s[tid] = 0.f;
  __syncthreads();

  const size_t bbase = (size_t)b * TT * UU;
  for (int t = 0; t < TT; ++t) {
    const size_t sbase = bbase + (size_t)t * UU;
    if (tid < 256) {
      int u = tid;
      float iv = ig[sbase + u];
      float fv = fg[sbase + u];
      float vv = vg[sbase + u];
      float kv = kg[sbase + u];
      float qv = qg[sbase + u];
      float ov = og[sbase + u];
      f_s[u] = fv;
      a_s[u] = iv * vv;
      k_s[u] = kv;
      q_s[u] = qv;
      o_s[u] = 1.0f / (1.0f + __expf(-ov));
      float nn = fmaf(fv, n_s[u], iv * kv);
      n_s[u] = nn;
      nq_s[u] = nn * qv;
    }
    __syncthreads();
    if (tid < 32) {  // wave0: denominator reduction while others update C
      float s = 0.f;
#pragma unroll
      for (int j = 0; j < 8; ++j) s += nq_s[tid + 32 * j];
      s = wave_reduce_sum(s);
      if (tid == 0) denom_s = fmaxf(fabsf(s), 1.0f);
    }
    float kr[8], qr[8];
#pragma unroll
    for (int j = 0; j < 8; ++j) {
      kr[j] = k_s[l + 32 * j];  // conflict-free: 32 consecutive banks
      qr[j] = q_s[l + 32 * j];
    }
#pragma unroll
    for (int i = 0; i < 8; ++i) {
      int r = w + 32 * i;
      float fr = f_s[r], ar = a_s[r];  // broadcast reads
      float acc = 0.f;
#pragma unroll
      for (int j = 0; j < 8; ++j) {
        float c = fmaf(fr, C[i][j], ar * kr[j]);  // C_t = f*C + (i*v) k^T
        C[i][j] = c;
        acc = fmaf(c, qr[j], acc);  // numerator partial
      }
      acc = wave_reduce_sum(acc);
      if (l == 0) num_s[r] = acc;
    }
    __syncthreads();
    if (tid < 256) out[sbase + tid] = o_s[tid] * num_s[tid] / denom_s;
    __syncthreads();  // protect LDS before next step's staging
  }
}

extern "C" void kernel_launch(void* const* d_in, const int* in_sizes, int n_in,
                              void* d_out, int out_size, void* d_ws, size_t ws_size,
                              hipStream_t stream) {
  const float* x     = (const float*)d_in[0];
  const float* gamma = (const float*)d_in[1];
  const float* beta  = (const float*)d_in[2];
  const float* Wq = (const float*)d_in[3];  const float* bq = (const float*)d_in[4];
  const float* Wk = (const float*)d_in[5];  const float* bk = (const float*)d_in[6];
  const float* Wv = (const float*)d_in[7];  const float* bv = (const float*)d_in[8];
  const float* Wi = (const float*)d_in[9];  const float* bi = (const float*)d_in[10];
  const float* Wf = (const float*)d_in[11]; const float* bf = (const float*)d_in[12];
  const float* Wo = (const float*)d_in[13]; const float* bo = (const float*)d_in[14];
  float* out = (float*)d_out;

  const size_t NE = (size_t)BB * TT * UU;  // 1,048,576 elems per activation
  float* ws = (float*)d_ws;
  float* xn = ws;
  float* aq = xn + NE;
  float* ak = aq + NE;
  float* av = ak + NE;
  float* ai = av + NE;
  float* af = ai + NE;
  float* ao = af + NE;

  ln_kernel<<<BB * TT / 8, 256, 0, stream>>>(x, gamma, beta, xn);

  dim3 ggrid(UU / 128, BB * TT / 32);  // (2, 128)
  gemm_kernel<<<ggrid, 256, 0, stream>>>(x,  Wq, bq, aq, 1.0f);
  gemm_kernel<<<ggrid, 256, 0, stream>>>(xn, Wk, bk, ak, 0.0625f);  // * 1/sqrt(256)
  gemm_kernel<<<ggrid, 256, 0, stream>>>(xn, Wv, bv, av, 1.0f);
  gemm_kernel<<<ggrid, 256, 0, stream>>>(x,  Wi, bi, ai, 1.0f);
  gemm_kernel<<<ggrid, 256, 0, stream>>>(x,  Wf, bf, af, 1.0f);
  gemm_kernel<<<ggrid, 256, 0, stream>>>(x,  Wo, bo, ao, 1.0f);

  gate_kernel<<<BB * UU / 256, 256, 0, stream>>>(ai, af);

  recur_kernel<<<BB, 1024, 0, stream>>>(aq, ak, av, ai, af, ao, out);
}